// SSM_Layer_1228360646847
// MI455X (gfx1250) — compile-verified
//
#include <hip/hip_runtime.h>
#include <hip/hip_bf16.h>

#define Hh 128
#define Nn 64
#define Ll 4096
#define Bb 8

// ---------- complex helpers (complex64 as two floats) ----------
struct c64 { float re, im; };
__device__ __forceinline__ c64 cadd(c64 a, c64 b) { return {a.re + b.re, a.im + b.im}; }
__device__ __forceinline__ c64 csub(c64 a, c64 b) { return {a.re - b.re, a.im - b.im}; }
__device__ __forceinline__ c64 cmul(c64 a, c64 b) {
    return {a.re * b.re - a.im * b.im, a.re * b.im + a.im * b.re};
}
__device__ __forceinline__ c64 cinv(c64 a) {
    float id = 1.0f / (a.re * a.re + a.im * a.im);
    return {a.re * id, -a.im * id};
}
__device__ __forceinline__ c64 cscale(float s, c64 a) { return {s * a.re, s * a.im}; }
__device__ __forceinline__ c64 cfma(c64 a, c64 b, c64 acc) {
    return {fmaf(a.re, b.re, fmaf(-a.im, b.im, acc.re)),
            fmaf(a.re, b.im, fmaf(a.im, b.re, acc.im))};
}

// =====================================================================
// Kernel 1: per-head SSM kernel generation.
//   Stage A: Cauchy resolvent -> at_roots[l] in LDS (complex64).
//   Stage B: in-place radix-2 DIF inverse FFT (sign +), bit-reversed
//            output unscrambled during the final store of Re(.)/L.
// One workgroup (512 threads = 16 wave32) per head h.
// =====================================================================
__global__ __launch_bounds__(512) void ssm_gen_kernel(
    const float* __restrict__ B_ri, const float* __restrict__ C_ri,
    const float* __restrict__ Lam_ri, const float* __restrict__ P_ri,
    const float* __restrict__ step_p, float* __restrict__ kern /* [H][L] */) {
    __shared__ c64 buf[Ll];                      // 32 KB
    __shared__ c64 lam_s[Nn];
    __shared__ c64 w00[Nn], w01[Nn], w10[Nn], w11[Nn];

    const int h   = blockIdx.x;
    const int tid = threadIdx.x;

    // per-head reduction vectors
    if (tid < Nn) {
        const int n    = tid;
        const int base = (h * Nn + n) * 2;
        c64 Bc = {B_ri[base], B_ri[base + 1]};
        c64 Cc = {C_ri[base], C_ri[base + 1]};
        c64 Pc = {P_ri[base], P_ri[base + 1]};
        lam_s[n] = {Lam_ri[base], Lam_ri[base + 1]};
        c64 Ccj = {Cc.re, -Cc.im};
        c64 Pcj = {Pc.re, -Pc.im};
        w00[n] = cmul(Ccj, Bc);   // conj(C)*B
        w01[n] = cmul(Ccj, Pc);   // conj(C)*P
        w10[n] = cmul(Pcj, Bc);   // conj(P)*B
        w11[n] = cmul(Pcj, Pc);   // conj(P)*P
    }
    __syncthreads();

    const float two_over_step = 2.0f / step_p[0];
    const float thstep = -6.28318530717958647692f / (float)Ll;

    // Stage A: at_roots into buf
    for (int l = tid; l < Ll; l += 512) {
        float zs, zc;
        __sincosf(thstep * (float)l, &zs, &zc);      // z = exp(-2*pi*i*l/L)
        c64 opz   = {1.0f + zc, zs};                 // 1 + z
        c64 iopz  = cinv(opz);
        c64 cc    = cscale(2.0f, iopz);              // c = 2/(1+z)
        c64 onemz = {1.0f - zc, -zs};                // 1 - z
        c64 g     = cscale(two_over_step, cmul(onemz, iopz));
        c64 a00 = {0.f, 0.f}, a01 = {0.f, 0.f}, a10 = {0.f, 0.f}, a11 = {0.f, 0.f};
#pragma unroll 8
        for (int n = 0; n < Nn; ++n) {
            c64 lam = lam_s[n];
            c64 r   = cinv(c64{g.re - lam.re, g.im - lam.im});  // 1/(g - lambda)
            a00 = cfma(w00[n], r, a00);
            a01 = cfma(w01[n], r, a01);
            a10 = cfma(w10[n], r, a10);
            a11 = cfma(w11[n], r, a11);
        }
        c64 kd = cinv(c64{1.0f + a11.re, a11.im});
        c64 at = cmul(cc, csub(a00, cmul(cmul(a01, kd), a10)));
        buf[l] = at;
    }
    __syncthreads();

    // Stage B: in-place radix-2 DIF inverse FFT (twiddle sign +).
    int log2half = 11;
    for (int size = Ll; size >= 2; size >>= 1, --log2half) {
        const int half = size >> 1;
        const float ang = 6.28318530717958647692f / (float)size;
        for (int j = tid; j < (Ll >> 1); j += 512) {
            const int q   = j & (half - 1);
            const int blk = j >> log2half;
            const int i0  = (blk << (log2half + 1)) + q;
            const int i1  = i0 + half;
            c64 x0 = buf[i0];
            c64 x1 = buf[i1];
            buf[i0] = cadd(x0, x1);
            float sw, cw;
            __sincosf(ang * (float)q, &sw, &cw);
            buf[i1] = cmul(csub(x0, x1), c64{cw, sw});
        }
        __syncthreads();
    }

    // bit-reversed unscramble + 1/L scale, keep real part
    const float invL = 1.0f / (float)Ll;
    for (int l = tid; l < Ll; l += 512) {
        const int rl = (int)(__brev((unsigned)l) >> 20);  // 12-bit reversal
        kern[h * Ll + rl] = buf[l].re * invL;
    }
}

// =====================================================================
// Kernel 2: causal depthwise convolution y[b,t,h] = sum_j k[h,j]*u[b,t-j,h]
//           + D[h]*u[b,t,h], mapped onto V_WMMA_F32_16X16X4_F32.
//
// Blocked Toeplitz form: Y[r,i] = sum_d K_d[16x16] @ U_d[16x16],
//   K_d[r,c] = k[16d + r - c]   (A operand: shared across all columns)
//   U_d[c,i] = u[16(i-d) + c]   (B operand)
// One workgroup per (h,b); wave g (of 16) owns output group g
// (times [256g, 256g+256)) and sums lags d = 0..16g+15 (causal).
//
// ILP: the 4 K-slices (kk) of each 16x16x16 block product accumulate
// into 4 independent accumulators (K-partials commute), so consecutive
// WMMAs have no RAW dependence; summed once in the epilogue.
//
// LDS bank conflicts: staged u rows padded 16 -> 20 floats (UPAD).
// B-fragment lane stride becomes 20 dwords; 20*m mod 64 is distinct for
// m = 0..15 and each row contributes 4 consecutive dwords, so one b64
// read touches all 64 banks exactly once (conflict-free). Row starts
// stay 16B-aligned -> float2 reads stay 8B-aligned.
//
// Zero-padded prefixes (16 k-rows / 16 u-rows) make negative-lag
// indices branchless so EXEC stays all-ones around the WMMAs, and the
// d-loop bound is wave-uniform (g constant per wave): no divergence.
// =====================================================================
typedef __attribute__((ext_vector_type(2))) float v2f;
typedef __attribute__((ext_vector_type(8))) float v8f;

#define UPAD 20                       // floats per staged u row (16 data + 4 pad)
#define UROWS (16 + 256)              // 16 zero rows + 256 data rows
#define UZERO (16 * UPAD)             // phys floats in the zero prefix

__global__ __launch_bounds__(512) void ssm_conv_wmma(
    const float* __restrict__ u,    /* [B][L][H] */
    const float* __restrict__ kern, /* [H][L]    */
    const float* __restrict__ Dp,   /* [H][1]    */
    float* __restrict__ y /* [B][L][H] */) {
    __shared__ __align__(16) float lk[16 + Ll];        // 16-zero prefix for d=0 lower-tri
    __shared__ __align__(16) float lu[UROWS * UPAD];   // row-padded staged u

    const int h   = blockIdx.x;
    const int b   = blockIdx.y;
    const int tid = threadIdx.x;

    if (tid < 16) lk[tid] = 0.0f;
    if (tid < UZERO) lu[tid] = 0.0f;
    // contiguous kernel row: b128 copies (index 16+4i keeps 16B alignment)
    for (int i = tid * 4; i < Ll; i += 2048) {
        *(float4*)&lk[16 + i] = *(const float4*)&kern[h * Ll + i];
    }
    // strided u gather into padded rows: phys(t) = UZERO + 20*(t>>4) + (t&15)
    for (int i = tid; i < Ll; i += 512) {
        lu[UZERO + UPAD * (i >> 4) + (i & 15)] = u[((size_t)b * Ll + i) * Hh + h];
    }
    __syncthreads();

    const int lane = tid & 31;
    const int g    = tid >> 5;    // wave id = output group (uniform per wave)
    const int m    = lane & 15;   // A-row / B-col index
    const int kh   = lane >> 4;   // K-half selector

    v8f acc0 = {}, acc1 = {}, acc2 = {}, acc3 = {};
    const int dmax = 16 * g + 15;
    for (int d = 0; d <= dmax; ++d) {
        const int abase = 16 + 16 * d + m - 2 * kh;              // k index (descending in K)
        const int urow  = 16 * g + m - d + 16;                   // padded row (>= 1)
        const int ubase = UPAD * urow + 2 * kh;                  // u index (ascending in K)

        v2f a0; a0.x = lk[abase];      a0.y = lk[abase - 1];
        v2f a1; a1.x = lk[abase - 4];  a1.y = lk[abase - 5];
        v2f a2; a2.x = lk[abase - 8];  a2.y = lk[abase - 9];
        v2f a3; a3.x = lk[abase - 12]; a3.y = lk[abase - 13];
        v2f b0 = *(const v2f*)(&lu[ubase]);
        v2f b1 = *(const v2f*)(&lu[ubase + 4]);
        v2f b2 = *(const v2f*)(&lu[ubase + 8]);
        v2f b3 = *(const v2f*)(&lu[ubase + 12]);

        acc0 = __builtin_amdgcn_wmma_f32_16x16x4_f32(false, a0, false, b0, (short)0, acc0, false, false);
        acc1 = __builtin_amdgcn_wmma_f32_16x16x4_f32(false, a1, false, b1, (short)0, acc1, false, false);
        acc2 = __builtin_amdgcn_wmma_f32_16x16x4_f32(false, a2, false, b2, (short)0, acc2, false, false);
        acc3 = __builtin_amdgcn_wmma_f32_16x16x4_f32(false, a3, false, b3, (short)0, acc3, false, false);
    }
    v8f acc = (acc0 + acc1) + (acc2 + acc3);

    // epilogue: D skip + scatter to [B,L,H]
    const float dsk = Dp[h];
#pragma unroll
    for (int v = 0; v < 8; ++v) {
        const int r = v + 8 * kh;             // output row within tile
        const int t = 256 * g + 16 * m + r;   // absolute time
        y[((size_t)b * Ll + t) * Hh + h] = acc[v] + dsk * lu[UZERO + UPAD * (16 * g + m) + r];
    }
}

extern "C" void kernel_launch(void* const* d_in, const int* in_sizes, int n_in,
                              void* d_out, int out_size, void* d_ws, size_t ws_size,
                              hipStream_t stream) {
    (void)in_sizes; (void)n_in; (void)out_size; (void)ws_size;
    const float* u      = (const float*)d_in[0];  // [8, 4096, 128]
    const float* B_ri   = (const float*)d_in[1];  // [128, 64, 2]
    const float* C_ri   = (const float*)d_in[2];  // [128, 64, 2]
    const float* D      = (const float*)d_in[3];  // [128, 1]
    const float* Lam_ri = (const float*)d_in[4];  // [128, 64, 2]
    const float* P_ri   = (const float*)d_in[5];  // [128, 64, 2]
    const float* step   = (const float*)d_in[6];  // [1]
    float* y    = (float*)d_out;                  // [8, 4096, 128]
    float* kern = (float*)d_ws;                   // [128][4096] = 2 MB scratch

    ssm_gen_kernel<<<dim3(Hh), dim3(512), 0, stream>>>(B_ri, C_ri, Lam_ri, P_ri, step, kern);
    ssm_conv_wmma<<<dim3(Hh, Bb), dim3(512), 0, stream>>>(u, kern, D, y);
}